// EncoderLayer_8993661518580
// MI455X (gfx1250) — compile-verified
//
#include <hip/hip_runtime.h>

// ---------------------------------------------------------------------------
// SparseConvNet-style UNet on 128^3 grid for MI455X (gfx1250, wave32).
// fp32 end-to-end; middle conv stages run as GEMMs on v_wmma_f32_16x16x4_f32.
// All BN-apply+ReLU+mask work is fused into the conv input reads; BN stats
// are a single fused sum/sumsq/count reduction pass per stage.
// ---------------------------------------------------------------------------

typedef float v2f __attribute__((ext_vector_type(2)));
typedef float v8f __attribute__((ext_vector_type(8)));

constexpr long V0 = 128L * 128 * 128;
constexpr long V1 = 64L * 64 * 64;
constexpr long V2 = 32L * 32 * 32;
constexpr long V3 = 16L * 16 * 16;
constexpr long V4 = 8L * 8 * 8;
constexpr long V5 = 4L * 4 * 4;

// float-offsets into workspace
constexpr long OFF_GSUM = 0;                 // 128^3  (becomes averaged x0-in)
constexpr long OFF_GCNT = OFF_GSUM + V0;     // 128^3  (becomes mask0)
constexpr long OFF_X0   = OFF_GCNT + V0;     // 128^3 * 4
constexpr long OFF_X1   = OFF_X0 + 4 * V0;   // 64^3 * 8
constexpr long OFF_X2   = OFF_X1 + 8 * V1;   // 32^3 * 16
constexpr long OFF_X3   = OFF_X2 + 16 * V2;  // 16^3 * 32
constexpr long OFF_X4   = OFF_X3 + 32 * V3;  // 8^3 * 64
constexpr long OFF_X5   = OFF_X4 + 64 * V4;  // 4^3 * 128
constexpr long OFF_M1   = OFF_X5 + 128 * V5;
constexpr long OFF_M2   = OFF_M1 + V1;
constexpr long OFF_M3   = OFF_M2 + V2;
constexpr long OFF_M4   = OFF_M3 + V3;
constexpr long OFF_M5   = OFF_M4 + V4;
constexpr long OFF_Y4   = OFF_M5 + V5;       // 8^3 * 64
constexpr long OFF_Y3   = OFF_Y4 + 64 * V4;  // 16^3 * 32
constexpr long OFF_Y2   = OFF_Y3 + 32 * V3;  // 32^3 * 16
constexpr long OFF_Y1   = OFF_Y2 + 16 * V2;  // 64^3 * 8
constexpr long OFF_ST   = OFF_Y1 + 8 * V1;   // 10 stages * 1024 floats
constexpr long STATS_STRIDE = 1024;          // raw at +0 (2C+1 used), norm at +512 (3C used)

// ---------------------------------------------------------------------------
// utility kernels
// ---------------------------------------------------------------------------
__global__ void zero_f32(float* __restrict__ p, int n) {
  int i = blockIdx.x * 256 + threadIdx.x;
  if (i < n) p[i] = 0.0f;
}

__global__ void scatter_points(const int* __restrict__ coords,
                               const float* __restrict__ feats, int n,
                               float* __restrict__ gsum, float* __restrict__ gcnt) {
  int i = blockIdx.x * 256 + threadIdx.x;
  if (i >= n) return;
  int z = coords[3 * i + 0], y = coords[3 * i + 1], x = coords[3 * i + 2];
  int lin = (z * 128 + y) * 128 + x;
  atomicAdd(&gsum[lin], feats[i]);
  atomicAdd(&gcnt[lin], 1.0f);
}

// in-place: gsum -> averaged features, gcnt -> {0,1} mask
__global__ void voxel_average(float* __restrict__ gsum, float* __restrict__ gcnt) {
  int v = blockIdx.x * 256 + threadIdx.x;
  if (v >= (int)V0) return;
  float c = gcnt[v];
  gsum[v] = gsum[v] / fmaxf(c, 1.0f);
  gcnt[v] = (c > 0.0f) ? 1.0f : 0.0f;
}

// submanifold 3x3x3 conv, 1 -> 4 channels, SAME, masked to active sites
__global__ void subm_conv3(const float* __restrict__ xin, const float* __restrict__ mask,
                           const float* __restrict__ w, float* __restrict__ xout) {
  int v = blockIdx.x * 256 + threadIdx.x;
  if (v >= (int)V0) return;
  int x = v & 127, y = (v >> 7) & 127, z = v >> 14;
  float m = mask[v];
  float acc0 = 0.f, acc1 = 0.f, acc2 = 0.f, acc3 = 0.f;
#pragma unroll
  for (int kz = 0; kz < 3; kz++) {
    int iz = z + kz - 1;
    if ((unsigned)iz >= 128u) continue;
#pragma unroll
    for (int ky = 0; ky < 3; ky++) {
      int iy = y + ky - 1;
      if ((unsigned)iy >= 128u) continue;
#pragma unroll
      for (int kx = 0; kx < 3; kx++) {
        int ix = x + kx - 1;
        if ((unsigned)ix >= 128u) continue;
        float val = xin[(iz * 128 + iy) * 128 + ix];
        const float* wk = w + ((kz * 3 + ky) * 3 + kx) * 4;
        acc0 = fmaf(val, wk[0], acc0);
        acc1 = fmaf(val, wk[1], acc1);
        acc2 = fmaf(val, wk[2], acc2);
        acc3 = fmaf(val, wk[3], acc3);
      }
    }
  }
  float* o = xout + (long)v * 4;
  o[0] = acc0 * m; o[1] = acc1 * m; o[2] = acc2 * m; o[3] = acc3 * m;
}

// ---------------------------------------------------------------------------
// masked BatchNorm statistics: per-channel sum & sumsq (x is zero at inactive
// sites, so plain sums equal masked sums) + active-voxel count.
// layout channel-last; 256 % C == 0 so each thread owns a fixed channel.
// ---------------------------------------------------------------------------
template <int C>
__global__ void bn_stats(const float* __restrict__ x, const float* __restrict__ mask,
                         int nvox, float* __restrict__ raw) {
  __shared__ float sS[256];
  __shared__ float sQ[256];
  int tid = threadIdx.x;
  long stride = (long)gridDim.x * 256;
  long nelem = (long)nvox * C;
  float s = 0.f, q = 0.f;
  for (long idx = (long)blockIdx.x * 256 + tid; idx < nelem; idx += stride) {
    float v = x[idx];
    s += v;
    q += v * v;
  }
  sS[tid] = s; sQ[tid] = q;
  __syncthreads();
  for (int off = 128; off >= C; off >>= 1) {   // folds preserve tid mod C
    if (tid < off) { sS[tid] += sS[tid + off]; sQ[tid] += sQ[tid + off]; }
    __syncthreads();
  }
  if (tid < C) { atomicAdd(&raw[tid], sS[tid]); atomicAdd(&raw[C + tid], sQ[tid]); }
  __syncthreads();
  float cnt = 0.f;
  for (long v = (long)blockIdx.x * 256 + tid; v < nvox; v += stride) cnt += mask[v];
  sS[tid] = cnt;
  __syncthreads();
  for (int off = 128; off >= 1; off >>= 1) {
    if (tid < off) sS[tid] += sS[tid + off];
    __syncthreads();
  }
  if (tid == 0) atomicAdd(&raw[2 * C], sS[0]);
}

template <int C>
__global__ void bn_finalize(const float* __restrict__ raw, const float* __restrict__ gamma,
                            const float* __restrict__ beta, float* __restrict__ nb) {
  int c = threadIdx.x;
  if (c >= C) return;
  float cnt = fmaxf(raw[2 * C], 1.0f);
  float mean = raw[c] / cnt;
  float var = raw[C + c] / cnt - mean * mean;
  nb[c] = mean;
  nb[C + c] = rsqrtf(var + 1e-4f) * gamma[c];
  nb[2 * C + c] = beta[c];
}

__device__ __forceinline__ float bn_relu(float v, const float* nb, int C, int ci, float m) {
  v = (v - nb[ci]) * nb[C + ci] + nb[2 * C + ci];
  v = v > 0.f ? v : 0.f;   // LEAK == 0
  return v * m;
}

// ---------------------------------------------------------------------------
// plain fused encoder stage: BN+ReLU+mask -> k2s2 conv -> pooled mask
// (used for tiny-channel stages where the work is pure bandwidth)
// ---------------------------------------------------------------------------
template <int CIN, int COUT, int SOUT>
__global__ void enc_conv_plain(const float* __restrict__ xin, const float* __restrict__ maskin,
                               const float* __restrict__ w, const float* __restrict__ nb,
                               float* __restrict__ xout, float* __restrict__ maskout) {
  int vo = blockIdx.x * 256 + threadIdx.x;
  constexpr int NV = SOUT * SOUT * SOUT;
  if (vo >= NV) return;
  int xo = vo % SOUT, yo = (vo / SOUT) % SOUT, zo = vo / (SOUT * SOUT);
  constexpr int SIN = SOUT * 2;
  float acc[COUT];
#pragma unroll
  for (int c = 0; c < COUT; c++) acc[c] = 0.f;
  float mo = 0.f;
#pragma unroll
  for (int t = 0; t < 8; t++) {
    int pz = 2 * zo + (t >> 2), py = 2 * yo + ((t >> 1) & 1), px = 2 * xo + (t & 1);
    int pv = (pz * SIN + py) * SIN + px;
    float m = maskin[pv];
    mo = fmaxf(mo, m);
#pragma unroll
    for (int ci = 0; ci < CIN; ci++) {
      float v = bn_relu(xin[(long)pv * CIN + ci], nb, CIN, ci, m);
      const float* wk = w + (t * CIN + ci) * COUT;
#pragma unroll
      for (int co = 0; co < COUT; co++) acc[co] = fmaf(v, wk[co], acc[co]);
    }
  }
  maskout[vo] = mo;
#pragma unroll
  for (int co = 0; co < COUT; co++) xout[(long)vo * COUT + co] = acc[co] * mo;
}

// ---------------------------------------------------------------------------
// WMMA fused encoder stage as GEMM: M = out voxels, N = COUT, K = 8*CIN.
// 1 wave computes a 16x16 D tile with V_WMMA_F32_16X16X4_F32, K chunked
// through LDS (<=16KB/block).
//
// f32 16x4 A fragment layout (ISA 7.12.2): lanes 0-15 hold rows M=0..15,
// VGPR0 = {K=0 | K=2 for upper half-wave}, VGPR1 = {K=1 | K=3}.
// B (4x16) is the row-striped mirror: per lane col N=lane&15, half-wave
// selects K pair. D: VGPR r -> row r (lanes 0-15) / r+8 (lanes 16-31).
// ---------------------------------------------------------------------------
template <int CIN, int COUT, int SOUT>
__global__ void __launch_bounds__(32)
enc_conv_wmma(const float* __restrict__ xin, const float* __restrict__ maskin,
              const float* __restrict__ w, const float* __restrict__ nb,
              float* __restrict__ xout, float* __restrict__ maskout) {
  constexpr int K = 8 * CIN;
  constexpr int KC = (K < 128) ? K : 128;
  __shared__ float As[16 * KC];
  __shared__ float Bs[KC * 16];
  __shared__ float Ms[16];
  const int lane = threadIdx.x;
  const int mBase = blockIdx.x * 16;
  const int nBase = blockIdx.y * 16;
  constexpr int SIN = SOUT * 2;

  if (lane < 16) {  // pooled mask for this row tile
    int vo = mBase + lane;
    int xo = vo % SOUT, yo = (vo / SOUT) % SOUT, zo = vo / (SOUT * SOUT);
    float mo = 0.f;
#pragma unroll
    for (int t = 0; t < 8; t++) {
      int pv = ((2 * zo + (t >> 2)) * SIN + (2 * yo + ((t >> 1) & 1))) * SIN + (2 * xo + (t & 1));
      mo = fmaxf(mo, maskin[pv]);
    }
    Ms[lane] = mo;
    maskout[vo] = mo;
  }

  v8f acc = {};
  const int half = lane >> 4;
  const int mrow = lane & 15;
  const int ncol = lane & 15;

  for (int kb = 0; kb < K; kb += KC) {
    // stage A chunk: BN+ReLU+mask applied on the fly
    for (int e = lane; e < 16 * KC; e += 32) {
      int m = e / KC, kk = e % KC, k = kb + kk;
      int t = k / CIN, ci = k % CIN;
      int vo = mBase + m;
      int xo = vo % SOUT, yo = (vo / SOUT) % SOUT, zo = vo / (SOUT * SOUT);
      int pv = ((2 * zo + (t >> 2)) * SIN + (2 * yo + ((t >> 1) & 1))) * SIN + (2 * xo + (t & 1));
      As[e] = bn_relu(xin[(long)pv * CIN + ci], nb, CIN, ci, maskin[pv]);
    }
    // stage B chunk: W[(t*CIN+ci)*COUT + co] == W[k*COUT + co]
    for (int e = lane; e < KC * 16; e += 32) {
      int kk = e >> 4, n = e & 15;
      int co = nBase + n;
      Bs[e] = (co < COUT) ? w[(kb + kk) * COUT + co] : 0.f;
    }
    __syncthreads();
#pragma unroll 4
    for (int kk = 0; kk < KC; kk += 4) {
      v2f a, b;
      a.x = As[mrow * KC + kk + 2 * half];
      a.y = As[mrow * KC + kk + 2 * half + 1];
      b.x = Bs[(kk + 2 * half) * 16 + ncol];
      b.y = Bs[(kk + 2 * half + 1) * 16 + ncol];
      acc = __builtin_amdgcn_wmma_f32_16x16x4_f32(false, a, false, b, (short)0, acc,
                                                  false, false);
    }
    __syncthreads();
  }

#pragma unroll
  for (int r = 0; r < 8; r++) {
    int m = r + half * 8;
    int vo = mBase + m;
    int co = nBase + ncol;
    if (co < COUT) xout[(long)vo * COUT + co] = acc[r] * Ms[m];
  }
}

// ---------------------------------------------------------------------------
// WMMA fused decoder stage (k2s2 transpose conv) as GEMM:
// M = coarse voxels, K = C1, N = 8*C0 (tap folded into N).
// D[m, t*C0+co] -> fine voxel (2*coords(m)+t), masked by fine mask.
// ---------------------------------------------------------------------------
template <int C1, int C0, int SC>
__global__ void __launch_bounds__(32)
dec_conv_wmma(const float* __restrict__ yin, const float* __restrict__ maskC,
              const float* __restrict__ maskF, const float* __restrict__ w,
              const float* __restrict__ nb, float* __restrict__ yout) {
  constexpr int K = C1;
  constexpr int KC = (K < 128) ? K : 128;
  __shared__ float As[16 * KC];
  __shared__ float Bs[KC * 16];
  const int lane = threadIdx.x;
  const int mBase = blockIdx.x * 16;
  const int nBase = blockIdx.y * 16;
  constexpr int SF = 2 * SC;

  v8f acc = {};
  const int half = lane >> 4;
  const int mrow = lane & 15;
  const int ncol = lane & 15;

  for (int kb = 0; kb < K; kb += KC) {
    for (int e = lane; e < 16 * KC; e += 32) {
      int m = e / KC, kk = e % KC, ci = kb + kk;
      int v = mBase + m;
      As[e] = bn_relu(yin[(long)v * C1 + ci], nb, C1, ci, maskC[v]);
    }
    for (int e = lane; e < KC * 16; e += 32) {
      int kk = e >> 4, n = e & 15;
      int ng = nBase + n;            // grid sized so ng < 8*C0 always
      int t = ng / C0, co = ng % C0;
      Bs[e] = w[(t * C1 + (kb + kk)) * C0 + co];
    }
    __syncthreads();
#pragma unroll 4
    for (int kk = 0; kk < KC; kk += 4) {
      v2f a, b;
      a.x = As[mrow * KC + kk + 2 * half];
      a.y = As[mrow * KC + kk + 2 * half + 1];
      b.x = Bs[(kk + 2 * half) * 16 + ncol];
      b.y = Bs[(kk + 2 * half + 1) * 16 + ncol];
      acc = __builtin_amdgcn_wmma_f32_16x16x4_f32(false, a, false, b, (short)0, acc,
                                                  false, false);
    }
    __syncthreads();
  }

#pragma unroll
  for (int r = 0; r < 8; r++) {
    int m = r + half * 8;
    int v = mBase + m;
    int xq = v % SC, yq = (v / SC) % SC, zq = v / (SC * SC);
    int ng = nBase + ncol;
    int t = ng / C0, co = ng % C0;
    int f = ((2 * zq + (t >> 2)) * SF + (2 * yq + ((t >> 1) & 1))) * SF + (2 * xq + (t & 1));
    yout[(long)f * C0 + co] = acc[r] * maskF[f];
  }
}

// plain fused decoder stage for the bandwidth-bound fine levels
template <int C1, int C0, int SF>
__global__ void dec_conv_plain(const float* __restrict__ yin, const float* __restrict__ maskC,
                               const float* __restrict__ maskF, const float* __restrict__ w,
                               const float* __restrict__ nb, float* __restrict__ yout) {
  int f = blockIdx.x * 256 + threadIdx.x;
  constexpr int NF = SF * SF * SF;
  if (f >= NF) return;
  int fx = f % SF, fy = (f / SF) % SF, fz = f / (SF * SF);
  constexpr int SC = SF / 2;
  int t = ((fz & 1) << 2) | ((fy & 1) << 1) | (fx & 1);
  int v = ((fz >> 1) * SC + (fy >> 1)) * SC + (fx >> 1);
  float mC = maskC[v];
  float mF = maskF[f];
  float acc[C0];
#pragma unroll
  for (int c = 0; c < C0; c++) acc[c] = 0.f;
#pragma unroll
  for (int ci = 0; ci < C1; ci++) {
    float val = bn_relu(yin[(long)v * C1 + ci], nb, C1, ci, mC);
    const float* wk = w + (t * C1 + ci) * C0;
#pragma unroll
    for (int co = 0; co < C0; co++) acc[co] = fmaf(val, wk[co], acc[co]);
  }
#pragma unroll
  for (int co = 0; co < C0; co++) yout[(long)f * C0 + co] = acc[co] * mF;
}

// hidden = NCDHW flatten of 4^3 x 128 bottleneck
__global__ void hidden_flatten(const float* __restrict__ x5, float* __restrict__ out) {
  int i = blockIdx.x * 256 + threadIdx.x;
  if (i >= 8192) return;
  int c = i >> 6, v = i & 63;
  out[i] = x5[v * 128 + c];
}

// ---------------------------------------------------------------------------
// host driver
// ---------------------------------------------------------------------------
extern "C" void kernel_launch(void* const* d_in, const int* in_sizes, int n_in,
                              void* d_out, int out_size, void* d_ws, size_t ws_size,
                              hipStream_t stream) {
  (void)n_in; (void)out_size; (void)ws_size;
  const int* coords = (const int*)d_in[0];
  const float* feats = (const float*)d_in[1];
  const float* w_sub = (const float*)d_in[2];
  const float *enc_w[5], *enc_g[5], *enc_b[5], *dec_w[5], *dec_g[5], *dec_b[5];
  for (int i = 0; i < 5; i++) {
    enc_w[i] = (const float*)d_in[3 + i];
    enc_g[i] = (const float*)d_in[8 + i];
    enc_b[i] = (const float*)d_in[13 + i];
    dec_w[i] = (const float*)d_in[18 + i];
    dec_g[i] = (const float*)d_in[23 + i];
    dec_b[i] = (const float*)d_in[28 + i];
  }
  const int npts = in_sizes[1];

  float* ws = (float*)d_ws;
  float* gsum = ws + OFF_GSUM;   // becomes x0-in
  float* gcnt = ws + OFF_GCNT;   // becomes mask0
  float* x0 = ws + OFF_X0;
  float* x1 = ws + OFF_X1;
  float* x2 = ws + OFF_X2;
  float* x3 = ws + OFF_X3;
  float* x4 = ws + OFF_X4;
  float* x5 = ws + OFF_X5;
  float* m1 = ws + OFF_M1;
  float* m2 = ws + OFF_M2;
  float* m3 = ws + OFF_M3;
  float* m4 = ws + OFF_M4;
  float* m5 = ws + OFF_M5;
  float* y4 = ws + OFF_Y4;
  float* y3 = ws + OFF_Y3;
  float* y2 = ws + OFF_Y2;
  float* y1 = ws + OFF_Y1;
  float* stats = ws + OFF_ST;
  float* out_x = (float*)d_out;             // [1,128,128,128,4] NDHWC flat
  float* out_h = (float*)d_out + 4 * V0;    // [1,8192]

#define NBLK(n) ((int)(((long)(n) + 255) / 256))
#define RUN_STATS(C, xptr, mptr, nvox, sidx, gma, bta)                                   \
  do {                                                                                   \
    float* raw_ = stats + (sidx)*STATS_STRIDE;                                           \
    int b_ = NBLK((long)(nvox) * (C));                                                   \
    if (b_ > 768) b_ = 768;                                                              \
    bn_stats<C><<<b_, 256, 0, stream>>>(xptr, mptr, (int)(nvox), raw_);                  \
    bn_finalize<C><<<1, 128, 0, stream>>>(raw_, gma, bta, raw_ + 512);                   \
  } while (0)
#define NB(sidx) (stats + (sidx)*STATS_STRIDE + 512)

  // 0) zero accumulators (gsum+gcnt contiguous) and all BN stat buffers
  zero_f32<<<NBLK(2 * V0), 256, 0, stream>>>(gsum, (int)(2 * V0));
  zero_f32<<<NBLK(10 * STATS_STRIDE), 256, 0, stream>>>(stats, (int)(10 * STATS_STRIDE));

  // 1) scatter-average points onto the voxel grid, build mask0
  scatter_points<<<NBLK(npts), 256, 0, stream>>>(coords, feats, npts, gsum, gcnt);
  voxel_average<<<NBLK(V0), 256, 0, stream>>>(gsum, gcnt);

  // 2) submanifold 3^3 conv 1->4
  subm_conv3<<<NBLK(V0), 256, 0, stream>>>(gsum, gcnt, w_sub, x0);

  // 3) encoder
  RUN_STATS(4, x0, gcnt, V0, 0, enc_g[0], enc_b[0]);
  enc_conv_plain<4, 8, 64><<<NBLK(V1), 256, 0, stream>>>(x0, gcnt, enc_w[0], NB(0), x1, m1);

  RUN_STATS(8, x1, m1, V1, 1, enc_g[1], enc_b[1]);
  enc_conv_plain<8, 16, 32><<<NBLK(V2), 256, 0, stream>>>(x1, m1, enc_w[1], NB(1), x2, m2);

  RUN_STATS(16, x2, m2, V2, 2, enc_g[2], enc_b[2]);
  enc_conv_wmma<16, 32, 16><<<dim3(V3 / 16, 2), 32, 0, stream>>>(x2, m2, enc_w[2], NB(2), x3, m3);

  RUN_STATS(32, x3, m3, V3, 3, enc_g[3], enc_b[3]);
  enc_conv_wmma<32, 64, 8><<<dim3(V4 / 16, 4), 32, 0, stream>>>(x3, m3, enc_w[3], NB(3), x4, m4);

  RUN_STATS(64, x4, m4, V4, 4, enc_g[4], enc_b[4]);
  enc_conv_wmma<64, 128, 4><<<dim3(V5 / 16, 8), 32, 0, stream>>>(x4, m4, enc_w[4], NB(4), x5, m5);

  // 4) hidden output (NCDHW flatten of bottleneck)
  hidden_flatten<<<NBLK(8192), 256, 0, stream>>>(x5, out_h);

  // 5) decoder (no skip connections; masks restore each finer active set)
  RUN_STATS(128, x5, m5, V5, 5, dec_g[0], dec_b[0]);
  dec_conv_wmma<128, 64, 4><<<dim3(V5 / 16, 32), 32, 0, stream>>>(x5, m5, m4, dec_w[0], NB(5), y4);

  RUN_STATS(64, y4, m4, V4, 6, dec_g[1], dec_b[1]);
  dec_conv_wmma<64, 32, 8><<<dim3(V4 / 16, 16), 32, 0, stream>>>(y4, m4, m3, dec_w[1], NB(6), y3);

  RUN_STATS(32, y3, m3, V3, 7, dec_g[2], dec_b[2]);
  dec_conv_wmma<32, 16, 16><<<dim3(V3 / 16, 8), 32, 0, stream>>>(y3, m3, m2, dec_w[2], NB(7), y2);

  RUN_STATS(16, y2, m2, V2, 8, dec_g[3], dec_b[3]);
  dec_conv_plain<16, 8, 64><<<NBLK(V1), 256, 0, stream>>>(y2, m2, m1, dec_w[3], NB(8), y1);

  RUN_STATS(8, y1, m1, V1, 9, dec_g[4], dec_b[4]);
  dec_conv_plain<8, 4, 128><<<NBLK(V0), 256, 0, stream>>>(y1, m1, gcnt, dec_w[4], NB(9), out_x);

#undef NB
#undef RUN_STATS
#undef NBLK
}